// HANLayer_36679020708090
// MI455X (gfx1250) — compile-verified
//
#include <hip/hip_runtime.h>
#include <hip/hip_bf16.h>
#include <math.h>

typedef __attribute__((ext_vector_type(2))) float v2f;
typedef __attribute__((ext_vector_type(8))) float v8f;
typedef __attribute__((ext_vector_type(4))) unsigned int v4u;
typedef __attribute__((ext_vector_type(4))) int v4i;
typedef __attribute__((ext_vector_type(8))) int v8i;

#define ROWP 68   // LDS row stride in floats: 64 data + 4 pad (bank-conflict-free)

// ---------------------------------------------------------------------------
// ws layout (floats):
//   zbuf : N*192          interleaved z[n][p][d], p=0 ident, p=1 e1, p=2 e2
//   deg1 : N              (later holds 1/deg)
//   deg2 : N
//   wsum : 3              semantic-attention pre-softmax accumulators
//   beta : 3              softmax weights
// ---------------------------------------------------------------------------

__device__ __forceinline__ void atomic_add_f32(float* p, float v) {
    unsafeAtomicAdd(p, v);   // global_atomic_add_f32 on gfx1250
}

// TDM: DMA a 16-row x 64-col f32 tile (row stride = stride_elems) from global
// into LDS with 4-DWORD row padding (LDS row stride = 68 floats). Called by
// one wave per block; completion enforced with s_wait_tensorcnt 0.
__device__ __forceinline__ void tdm_load_tile_16x64(const float* gsrc, void* lds_dst,
                                                    unsigned stride_elems) {
#if __has_builtin(__builtin_amdgcn_tensor_load_to_lds)
    unsigned long long ga = (unsigned long long)gsrc;
    unsigned lds_off = (unsigned)(unsigned long long)lds_dst;  // low 32 bits = LDS byte offset
    // D# group0: count=1 | lds_addr | global_addr[56:0] | type=2
    v4u g0 = { 1u,
               lds_off,
               (unsigned)ga,
               (unsigned)((ga >> 32) & 0x01FFFFFFu) | 0x80000000u };
    // D# group1: data_size=4B, pad_enable, pad_interval=64 DW (code 5),
    // pad_amount=4 DW (code 3); tensor_dim0=64, tensor_dim1=16,
    // tile_dim0=64, tile_dim1=16, tile_dim2=0, dim0_stride=stride_elems.
    v8i g1 = { (int)((2u << 16) | (1u << 20) | (5u << 22) | (3u << 25)),
               (int)(64u << 16),     // tensor_dim0[15:0] into bits 63:48
               (int)(16u << 16),     // tensor_dim0 hi=0 | tensor_dim1[15:0]
               (int)(64u << 16),     // tensor_dim1 hi=0 | tile_dim0=64
               16,                   // tile_dim1=16 | tile_dim2=0
               (int)stride_elems,    // tensor_dim0_stride lo32
               0, 0 };
    v4i zz4 = { 0, 0, 0, 0 };
    v8i zz8 = { 0, 0, 0, 0, 0, 0, 0, 0 };
    __builtin_amdgcn_tensor_load_to_lds(g0, g1, zz4, zz4, zz8, 0);
    __builtin_amdgcn_s_wait_tensorcnt(0);
#else
    // Fallback: cooperative load by the calling wave (32 lanes x float2).
    float* t = (float*)lds_dst;
    const int lane = threadIdx.x & 31;
    for (int r = 0; r < 16; ++r) {
        float2 v = *((const float2*)(gsrc + (long long)r * stride_elems) + lane);
        t[r * ROWP + lane * 2]     = v.x;
        t[r * ROWP + lane * 2 + 1] = v.y;
    }
#endif
}

// Zero z slots 1&2 (contiguous 128 floats per node at offset 64) + deg + wsum.
__global__ void k_zero(float* __restrict__ zbuf, float* __restrict__ degws,
                       long long nz, long long nd) {
    long long i = (long long)blockIdx.x * blockDim.x + threadIdx.x;
    if (i < nz) {
        long long n = i >> 7;
        int rem = (int)(i & 127);
        zbuf[n * 192 + 64 + rem] = 0.0f;
    } else if (i < nz + nd) {
        degws[i - nz] = 0.0f;
    }
}

// In-degree histograms for both meta-paths.
__global__ void k_degree(const int* __restrict__ dst1, const int* __restrict__ dst2,
                         float* __restrict__ deg1, float* __restrict__ deg2,
                         long long E) {
    long long i = (long long)blockIdx.x * blockDim.x + threadIdx.x;
    if (i < E) {
        atomic_add_f32(deg1 + dst1[i], 1.0f);
    } else if (i < 2 * E) {
        atomic_add_f32(deg2 + dst2[i - E], 1.0f);
    }
}

// deg -> 1/deg (0 if empty), in place over both arrays (contiguous 2N).
__global__ void k_invert(float* __restrict__ deg, long long n2) {
    long long i = (long long)blockIdx.x * blockDim.x + threadIdx.x;
    if (i < n2) {
        float v = deg[i];
        deg[i] = v > 0.0f ? 1.0f / v : 0.0f;
    }
}

// One wave per edge: gather h[src] (32 lanes x float2), scale by 1/deg[dst],
// atomic scatter-add into z slot p.
__global__ void k_scatter(const float* __restrict__ h,
                          const int* __restrict__ src1, const int* __restrict__ dst1,
                          const int* __restrict__ src2, const int* __restrict__ dst2,
                          const float* __restrict__ inv1, const float* __restrict__ inv2,
                          float* zbuf, long long E) {
    int lane = threadIdx.x & 31;
    long long eidx = ((long long)blockIdx.x * blockDim.x + threadIdx.x) >> 5;
    if (eidx >= 2 * E) return;
    int srcn, dstn, p;
    const float* inv;
    if (eidx < E) {
        srcn = src1[eidx]; dstn = dst1[eidx]; inv = inv1; p = 1;
    } else {
        long long e = eidx - E;
        srcn = src2[e]; dstn = dst2[e]; inv = inv2; p = 2;
    }
    float w = inv[dstn];
    float2 hv = *((const float2*)(h + (long long)srcn * 64) + lane);
    float* o = zbuf + (long long)dstn * 192 + p * 64 + lane * 2;
    atomic_add_f32(o,     hv.x * w);
    atomic_add_f32(o + 1, hv.y * w);
}

// Out[row,col] = act(A[row,:] @ W[:,col] + bias[col]), 64-wide GEMM via
// V_WMMA_F32_16X16X4_F32. 128 threads = 4 waves; wave w owns cols [16w,16w+16).
// A tile staged once per block into LDS by TDM (wave 0), published by the
// workgroup barrier; safe in place (A==Out) since the only global A read is
// the TDM before the barrier and all stores are after it.
__global__ void k_gemm64(const float* A, long long a_stride,
                         const float* __restrict__ W,
                         const float* __restrict__ bias,
                         float* Out, long long o_stride,
                         int do_elu) {
    __shared__ __align__(16) float tile[16 * ROWP];
    const int lane  = threadIdx.x & 31;
    const int wave  = threadIdx.x >> 5;
    const int row0  = blockIdx.x * 16;
    const int col0  = wave * 16;
    const int r     = lane & 15;
    const int khalf = lane >> 4;

    if (wave == 0)
        tdm_load_tile_16x64(A + (long long)row0 * a_stride, tile, (unsigned)a_stride);
    __syncthreads();

    v8f c = {};
#pragma unroll
    for (int k = 0; k < 16; ++k) {
        int kb = 4 * k + 2 * khalf;
        v2f a = *(const v2f*)&tile[r * ROWP + kb];          // ds_load_b64, conflict-free
        v2f bf;
        bf.x = W[(long long)kb * 64 + col0 + r];
        bf.y = W[(long long)(kb + 1) * 64 + col0 + r];
        c = __builtin_amdgcn_wmma_f32_16x16x4_f32(false, a, false, bf,
                                                  (short)0, c, false, false);
    }

    const int col = col0 + r;
    const float bv = bias[col];
#pragma unroll
    for (int v = 0; v < 8; ++v) {
        int rr = v + khalf * 8;       // C layout: lanes>=16 hold M=v+8
        float val = c[v] + bv;
        if (do_elu) val = val > 0.0f ? val : expm1f(val);
        Out[(long long)(row0 + rr) * o_stride + col] = val;
    }
}

// Semantic attention logits: sum over all nodes of tanh(z[:,p,:]@Wp1+bp1)@Wp2
// into wsum[p]. 256 threads = 8 waves; wave w owns hidden cols [16w,16w+16).
// z tile staged once per block via TDM (was 8x redundant global reads).
__global__ void k_attn(const float* __restrict__ Z,
                       const float* __restrict__ Wp1,   // [64,128]
                       const float* __restrict__ bp1,   // [128]
                       const float* __restrict__ Wp2,   // [128]
                       float* __restrict__ wsum) {
    __shared__ __align__(16) float tile[16 * ROWP];
    const int lane  = threadIdx.x & 31;
    const int wave  = threadIdx.x >> 5;
    const int p     = blockIdx.y;
    const int row0  = blockIdx.x * 16;
    const int col0  = wave * 16;
    const int r     = lane & 15;
    const int khalf = lane >> 4;

    if (wave == 0)
        tdm_load_tile_16x64(Z + (long long)p * 64 + (long long)row0 * 192, tile, 192u);
    __syncthreads();

    v8f c = {};
#pragma unroll
    for (int k = 0; k < 16; ++k) {
        int kb = 4 * k + 2 * khalf;
        v2f a = *(const v2f*)&tile[r * ROWP + kb];
        v2f b;
        b.x = Wp1[(long long)kb * 128 + col0 + r];
        b.y = Wp1[(long long)(kb + 1) * 128 + col0 + r];
        c = __builtin_amdgcn_wmma_f32_16x16x4_f32(false, a, false, b,
                                                  (short)0, c, false, false);
    }

    const int col = col0 + r;
    const float bv = bp1[col];
    const float w2 = Wp2[col];
    float part = 0.0f;
#pragma unroll
    for (int v = 0; v < 8; ++v)
        part += tanhf(c[v] + bv) * w2;   // mean over N => sum every element

    for (int off = 16; off > 0; off >>= 1)
        part += __shfl_xor(part, off, 32);
    if (lane == 0) atomic_add_f32(&wsum[p], part);
}

__global__ void k_softmax(const float* __restrict__ wsum,
                          float* __restrict__ beta, float invN) {
    float w0 = wsum[0] * invN, w1 = wsum[1] * invN, w2 = wsum[2] * invN;
    float m  = fmaxf(w0, fmaxf(w1, w2));
    float e0 = expf(w0 - m), e1 = expf(w1 - m), e2 = expf(w2 - m);
    float s  = e0 + e1 + e2;
    beta[0] = e0 / s; beta[1] = e1 / s; beta[2] = e2 / s;
}

__global__ void k_out(const float* __restrict__ Z, const float* __restrict__ beta,
                      float* __restrict__ out, long long total) {
    long long i = (long long)blockIdx.x * blockDim.x + threadIdx.x;
    if (i >= total) return;
    long long n = i >> 6;
    int d = (int)(i & 63);
    const float* zr = Z + n * 192 + d;
    out[i] = beta[0] * zr[0] + beta[1] * zr[64] + beta[2] * zr[128];
}

extern "C" void kernel_launch(void* const* d_in, const int* in_sizes, int n_in,
                              void* d_out, int out_size, void* d_ws, size_t ws_size,
                              hipStream_t stream) {
    const float* h    = (const float*)d_in[0];
    const int*   src1 = (const int*)  d_in[1];
    const int*   dst1 = (const int*)  d_in[2];
    const int*   src2 = (const int*)  d_in[3];
    const int*   dst2 = (const int*)  d_in[4];
    const float* W_id = (const float*)d_in[5];
    const float* b_id = (const float*)d_in[6];
    const float* W1   = (const float*)d_in[7];
    const float* b1   = (const float*)d_in[8];
    const float* W2   = (const float*)d_in[9];
    const float* b2   = (const float*)d_in[10];
    const float* Wp1  = (const float*)d_in[11];
    const float* bp1  = (const float*)d_in[12];
    const float* Wp2  = (const float*)d_in[13];

    const long long N = in_sizes[0] / 64;
    const long long E = in_sizes[1];

    float* zbuf = (float*)d_ws;          // N*192
    float* deg1 = zbuf + N * 192;        // N
    float* deg2 = deg1 + N;              // N
    float* wsum = deg2 + N;              // 3
    float* beta = wsum + 3;              // 3

    // 1. zero agg slots of z + deg1/deg2 + wsum
    {
        long long nz = N * 128, nd = 2 * N + 3, tot = nz + nd;
        k_zero<<<(unsigned)((tot + 255) / 256), 256, 0, stream>>>(zbuf, deg1, nz, nd);
    }
    // 2. in-degrees, then invert in place
    k_degree<<<(unsigned)((2 * E + 255) / 256), 256, 0, stream>>>(dst1, dst2, deg1, deg2, E);
    k_invert<<<(unsigned)((2 * N + 255) / 256), 256, 0, stream>>>(deg1, 2 * N);
    // 3. gather * (1/deg[dst]) -> atomic scatter-add into z slots 1,2
    k_scatter<<<(unsigned)((2 * E * 32 + 255) / 256), 256, 0, stream>>>(
        h, src1, dst1, src2, dst2, deg1, deg2, zbuf, E);
    // 4. projections (WMMA f32, TDM-staged A): ident into slot 0; slots 1,2 in place + ELU
    const unsigned rowtiles = (unsigned)(N / 16);
    k_gemm64<<<rowtiles, 128, 0, stream>>>(h,          64, W_id, b_id, zbuf,       192, 0);
    k_gemm64<<<rowtiles, 128, 0, stream>>>(zbuf + 64,  192, W1,   b1,  zbuf + 64,  192, 1);
    k_gemm64<<<rowtiles, 128, 0, stream>>>(zbuf + 128, 192, W2,   b2,  zbuf + 128, 192, 1);
    // 5. semantic attention logits (WMMA f32, TDM-staged z) -> softmax -> combine
    dim3 ag(rowtiles, 3);
    k_attn<<<ag, 256, 0, stream>>>(zbuf, Wp1, bp1, Wp2, wsum);
    k_softmax<<<1, 1, 0, stream>>>(wsum, beta, 1.0f / (float)N);
    k_out<<<(unsigned)((N * 64 + 255) / 256), 256, 0, stream>>>(zbuf, beta, (float*)d_out, N * 64);
}